// GraphIpaFrameDenoiser_7627861918030
// MI455X (gfx1250) — compile-verified
//
#include <hip/hip_runtime.h>
#include <hip/hip_bf16.h>
#include <math.h>

// ---------------- problem constants ----------------
#define NN    8192          // N_NODES
#define DEG   32
#define EE    (NN * DEG)    // 262144 edges
#define H     8
#define C     32
#define PQ    8
#define PV    12
#define CS    256
#define CZ    128

typedef float v2f __attribute__((ext_vector_type(2)));
typedef float v8f __attribute__((ext_vector_type(8)));

// =====================================================================
// Generic fp32 WMMA GEMM:  out[M,N] = A[M,K] @ W[K,N] (+bias)(+relu)
// block = 256 threads = 8 waves; block tile 32(M) x 64(N); K staged 32.
// Each wave computes one 16x16 tile with V_WMMA_F32_16X16X4_F32.
// Requires K % 32 == 0 (true for all call sites: 128,256,1664).
// Fast path: B128 (float4) staging when the tile is fully interior.
// =====================================================================
__global__ __launch_bounds__(256) void wmma_gemm_f32(
    const float* __restrict__ A, const float* __restrict__ W,
    const float* __restrict__ bias, float* __restrict__ out,
    int M, int K, int N, int relu)
{
    __shared__ __align__(16) float sA[32 * 32];
    __shared__ __align__(16) float sB[32 * 64];
    const int tid  = threadIdx.x;
    const int lane = tid & 31;
    const int wave = tid >> 5;
    const int wr = wave >> 2;          // 0..1  (16-row tile)
    const int wc = wave & 3;           // 0..3  (16-col tile)
    const int m0 = blockIdx.y * 32;
    const int n0 = blockIdx.x * 64;

    v8f acc = {};

    const int am  = wr * 16 + (lane & 15);
    const int bn  = wc * 16 + (lane & 15);
    const int kof = (lane >> 4) * 2;   // lanes 16-31 hold K+2,K+3

    const bool fullN = (n0 + 64 <= N) && ((N & 3) == 0); // uniform

    for (int k0 = 0; k0 < K; k0 += 32) {
        // ---- stage A tile 32x32: one float4 per thread (B128 loads) ----
        {
            int r = tid >> 3;               // 32 rows
            int c = (tid & 7) * 4;          // 8 float4 per row
            int gm = m0 + r;
            int gmc = (gm < M) ? gm : (M - 1);
            float4 av = *(const float4*)(A + (size_t)gmc * K + (k0 + c));
            if (gm >= M) av = make_float4(0.f, 0.f, 0.f, 0.f);
            *(float4*)&sA[r * 32 + c] = av;
        }
        // ---- stage B tile 32x64 ----
        if (fullN) {
#pragma unroll
            for (int i = 0; i < 2; ++i) {
                int idx = tid + i * 256;    // 512 float4
                int r = idx >> 4;           // 16 float4 per row
                int c = (idx & 15) * 4;
                *(float4*)&sB[r * 64 + c] =
                    *(const float4*)(W + (size_t)(k0 + r) * N + n0 + c);
            }
        } else {
            for (int idx = tid; idx < 32 * 64; idx += 256) {
                int r = idx >> 6, c = idx & 63;
                int gn = n0 + c;
                int gnc = (gn < N) ? gn : (N - 1);   // clamped, branch-free load
                float val = W[(size_t)(k0 + r) * N + gnc];
                sB[idx] = (gn < N) ? val : 0.f;
            }
        }
        // prefetch next weight slice into cache (global_prefetch_b8)
        if (k0 + 32 < K)
            __builtin_prefetch(&W[(size_t)(k0 + 32) * N + n0 + (tid & 63)], 0, 1);
        __syncthreads();

#pragma unroll
        for (int kk = 0; kk < 32; kk += 4) {
            v2f a, b;
            a.x = sA[am * 32 + kk + kof];
            a.y = sA[am * 32 + kk + kof + 1];
            b.x = sB[(kk + kof) * 64 + bn];
            b.y = sB[(kk + kof + 1) * 64 + bn];
            acc = __builtin_amdgcn_wmma_f32_16x16x4_f32(
                false, a, false, b, (short)0, acc, false, false);
        }
        __syncthreads();
    }

    const int gn = n0 + wc * 16 + (lane & 15);
    if (gn < N) {
        float bv = bias ? bias[gn] : 0.f;
#pragma unroll
        for (int r = 0; r < 8; ++r) {
            int gm = m0 + wr * 16 + (lane >> 4) * 8 + r;
            if (gm < M) {
                float v = acc[r] + bv;
                if (relu) v = fmaxf(v, 0.f);
                out[(size_t)gm * N + gn] = v;
            }
        }
    }
}

// =====================================================================
// Per-node quaternion normalize + rotation matrix
// =====================================================================
__global__ __launch_bounds__(256) void node_prep(
    const float* __restrict__ quats, float* __restrict__ qn,
    float* __restrict__ rot)
{
    int i = blockIdx.x * blockDim.x + threadIdx.x;
    if (i >= NN) return;
    float w = quats[i * 4 + 0], x = quats[i * 4 + 1];
    float y = quats[i * 4 + 2], z = quats[i * 4 + 3];
    float inv = rsqrtf(w * w + x * x + y * y + z * z);
    w *= inv; x *= inv; y *= inv; z *= inv;
    qn[i * 4 + 0] = w; qn[i * 4 + 1] = x; qn[i * 4 + 2] = y; qn[i * 4 + 3] = z;
    float* R = rot + (size_t)i * 9;
    R[0] = 1.f - 2.f * (y * y + z * z); R[1] = 2.f * (x * y - w * z); R[2] = 2.f * (x * z + w * y);
    R[3] = 2.f * (x * y + w * z); R[4] = 1.f - 2.f * (x * x + z * z); R[5] = 2.f * (y * z - w * x);
    R[6] = 2.f * (x * z - w * y); R[7] = 2.f * (y * z + w * x); R[8] = 1.f - 2.f * (x * x + y * y);
}

// in-place local->global point transform: p = R p + t  (P points per node)
__global__ __launch_bounds__(256) void transform_points(
    float* __restrict__ pts, const float* __restrict__ rot,
    const float* __restrict__ trans, int P)
{
    int i = blockIdx.x * blockDim.x + threadIdx.x;
    if (i >= NN * P) return;
    int n = i / P;
    float* p = pts + (size_t)i * 3;
    float x = p[0], y = p[1], z = p[2];
    const float* R = rot + (size_t)n * 9;
    const float* t = trans + (size_t)n * 3;
    p[0] = R[0] * x + R[1] * y + R[2] * z + t[0];
    p[1] = R[3] * x + R[4] * y + R[5] * z + t[1];
    p[2] = R[6] * x + R[7] * y + R[8] * z + t[2];
}

// =====================================================================
// Attention: one block per dst node; wave = head, lane = edge (wave32!)
// Segment softmax collapses to 32-lane shuffle reductions.
// All dot products use B128 loads (offsets are 16B aligned: 32*4, 24*4).
// =====================================================================
__global__ __launch_bounds__(256) void attn_kernel(
    const float* __restrict__ q, const float* __restrict__ k,
    const float* __restrict__ qg, const float* __restrict__ kg,
    const float* __restrict__ bws, const float* __restrict__ head_w,
    const float* __restrict__ rmask, const int* __restrict__ ei,
    float* __restrict__ attw)
{
    const int n = blockIdx.x;
    const int lane = threadIdx.x & 31;   // edge within node
    const int h = threadIdx.x >> 5;      // head
    const int e = n * DEG + lane;
    const int s = ei[e];                 // src node

    const float4* qp = (const float4*)(q + (size_t)n * (H * C) + h * C);
    const float4* kp = (const float4*)(k + (size_t)s * (H * C) + h * C);
    float dot = 0.f;
#pragma unroll
    for (int c = 0; c < C / 4; ++c) {
        float4 a = qp[c], b = kp[c];
        dot += a.x * b.x + a.y * b.y + a.z * b.z + a.w * b.w;
    }

    const float4* qgp = (const float4*)(qg + (size_t)n * (H * PQ * 3) + h * (PQ * 3));
    const float4* kgp = (const float4*)(kg + (size_t)s * (H * PQ * 3) + h * (PQ * 3));
    float pt = 0.f;
#pragma unroll
    for (int j = 0; j < (PQ * 3) / 4; ++j) {
        float4 a = qgp[j], b = kgp[j];
        float dx = a.x - b.x, dy = a.y - b.y, dz = a.z - b.z, dw = a.w - b.w;
        pt += dx * dx + dy * dy + dz * dz + dw * dw;
    }

    float hw = logf(1.f + expf(head_w[h])) * 0.09622504486493763f; // sqrt(1/108)
    float logit = dot * 0.10206207261596575f                        // sqrt(1/96)
                + 0.5773502691896258f * bws[(size_t)e * H + h]      // sqrt(1/3)
                - 0.5f * hw * pt
                + (rmask[s] - 1.f) * 100000.f;

    float mx = logit;
    for (int m = 16; m >= 1; m >>= 1) mx = fmaxf(mx, __shfl_xor(mx, m, 32));
    float ex = expf(logit - mx);
    float den = ex;
    for (int m = 16; m >= 1; m >>= 1) den += __shfl_xor(den, m, 32);
    attw[(size_t)e * H + h] = ex / (den + 1e-9f);
}

// =====================================================================
// Aggregation into the 1664-wide `cat` row; z tile staged in LDS (B128).
// cat layout: [o(256) | o_pt(288) | o_pt_norm(96) | o_pair(1024)]
// =====================================================================
__global__ __launch_bounds__(256) void aggregate_kernel(
    const float* __restrict__ v, const float* __restrict__ vg,
    const float* __restrict__ z, const float* __restrict__ attw,
    const int* __restrict__ ei, const float* __restrict__ rot,
    const float* __restrict__ trans, float* __restrict__ cat)
{
    __shared__ __align__(16) float zt[DEG * CZ];   // 16 KB
    __shared__ float att[DEG * H];
    __shared__ int srcs[DEG];
    const int n = blockIdx.x, tid = threadIdx.x;

    if (tid < DEG) srcs[tid] = ei[n * DEG + tid];
    { int e2 = tid >> 3, h = tid & 7;           // 256 == 32*8
      att[tid] = attw[(size_t)(n * DEG + e2) * H + h]; }
    // z tile: 1024 float4, 4 per thread
#pragma unroll
    for (int i = 0; i < 4; ++i) {
        int idx = tid + i * 256;
        int r = idx >> 5, c = (idx & 31) * 4;
        *(float4*)&zt[r * CZ + c] =
            *(const float4*)(z + (size_t)(n * DEG + r) * CZ + c);
    }
    __syncthreads();

    float* crow = cat + (size_t)n * 1664;

    // o : H*C = 256 outputs
    {
        int h = tid >> 5, c = tid & 31;
        float acc = 0.f;
        for (int e2 = 0; e2 < DEG; ++e2)
            acc += att[e2 * 8 + h] * v[(size_t)srcs[e2] * (H * C) + h * C + c];
        crow[h * C + c] = acc;
    }
    // o_pt & norm : 96 (h,p) pairs
    if (tid < H * PV) {
        int h = tid / PV, p = tid % PV;
        float gx = 0.f, gy = 0.f, gz = 0.f;
        for (int e2 = 0; e2 < DEG; ++e2) {
            float a = att[e2 * 8 + h];
            const float* vp = vg + (size_t)srcs[e2] * (H * PV * 3) + h * (PV * 3) + p * 3;
            gx += a * vp[0]; gy += a * vp[1]; gz += a * vp[2];
        }
        const float* R = rot + (size_t)n * 9;
        const float* t = trans + (size_t)n * 3;
        float dx = gx - t[0], dy = gy - t[1], dz = gz - t[2];
        float ox = R[0] * dx + R[3] * dy + R[6] * dz;  // R^T (einsum 'nji')
        float oy = R[1] * dx + R[4] * dy + R[7] * dz;
        float oz = R[2] * dx + R[5] * dy + R[8] * dz;
        crow[256 + h * 36 + p * 3 + 0] = ox;
        crow[256 + h * 36 + p * 3 + 1] = oy;
        crow[256 + h * 36 + p * 3 + 2] = oz;
        crow[256 + 288 + h * PV + p] = sqrtf(ox * ox + oy * oy + oz * oz + 1e-8f);
    }
    // o_pair : H*CZ = 1024 outputs
#pragma unroll
    for (int i = 0; i < 4; ++i) {
        int idx = tid + i * 256;
        int h = idx >> 7, c = idx & 127;
        float acc = 0.f;
        for (int e2 = 0; e2 < DEG; ++e2)
            acc += att[e2 * 8 + h] * zt[e2 * CZ + c];
        crow[640 + idx] = acc;
    }
}

// =====================================================================
// Fused residual + LayerNorm (D=256, one block per row)
//   y = LN(x + r*addmask) * postmask ; out2 = y * postmask2 (optional)
// =====================================================================
__global__ __launch_bounds__(256) void fused_ln(
    const float* __restrict__ x, const float* __restrict__ r,
    const float* __restrict__ addmask, const float* __restrict__ g,
    const float* __restrict__ bta, const float* __restrict__ postmask,
    const float* __restrict__ postmask2, float* __restrict__ outp,
    float* __restrict__ out2)
{
    __shared__ float red[256];
    __shared__ float sMu, sRs;
    const int n = blockIdx.x, t = threadIdx.x;
    const size_t base = (size_t)n * CS;
    float v = x[base + t];
    if (r) v += r[base + t] * (addmask ? addmask[n] : 1.f);
    red[t] = v; __syncthreads();
    for (int st = 128; st > 0; st >>= 1) { if (t < st) red[t] += red[t + st]; __syncthreads(); }
    if (t == 0) sMu = red[0] * (1.f / CS);
    __syncthreads();
    float d = v - sMu;
    red[t] = d * d; __syncthreads();
    for (int st = 128; st > 0; st >>= 1) { if (t < st) red[t] += red[t + st]; __syncthreads(); }
    if (t == 0) sRs = rsqrtf(red[0] * (1.f / CS) + 1e-5f);
    __syncthreads();
    float y = d * sRs * g[t] + bta[t];
    if (postmask) y *= postmask[n];
    outp[base + t] = y;
    if (out2) out2[base + t] = y * (postmask2 ? postmask2[n] : 1.f);
}

// =====================================================================
// Backbone update: quats & trans
// =====================================================================
__global__ __launch_bounds__(256) void bb_update(
    const float* __restrict__ upd, const float* __restrict__ nmask,
    const float* __restrict__ qn, const float* __restrict__ rot,
    const float* __restrict__ trans, float* __restrict__ qout,
    float* __restrict__ tout)
{
    int i = blockIdx.x * blockDim.x + threadIdx.x;
    if (i >= NN) return;
    float nm = nmask[i];
    float qx = upd[i * 6 + 0] * nm, qy = upd[i * 6 + 1] * nm, qz = upd[i * 6 + 2] * nm;
    float tx = upd[i * 6 + 3] * nm, ty = upd[i * 6 + 4] * nm, tz = upd[i * 6 + 5] * nm;
    float aw = qn[i * 4 + 0], ax = qn[i * 4 + 1], ay = qn[i * 4 + 2], az = qn[i * 4 + 3];
    // quat_mul(a, (1,qx,qy,qz))
    float nw = aw - ax * qx - ay * qy - az * qz;
    float nx = aw * qx + ax + ay * qz - az * qy;
    float ny = aw * qy - ax * qz + ay + az * qx;
    float nz = aw * qz + ax * qy - ay * qx + az;
    float inv = rsqrtf(nw * nw + nx * nx + ny * ny + nz * nz);
    qout[i * 4 + 0] = nw * inv; qout[i * 4 + 1] = nx * inv;
    qout[i * 4 + 2] = ny * inv; qout[i * 4 + 3] = nz * inv;
    const float* R = rot + (size_t)i * 9;
    tout[i * 3 + 0] = trans[i * 3 + 0] + R[0] * tx + R[1] * ty + R[2] * tz;
    tout[i * 3 + 1] = trans[i * 3 + 1] + R[3] * tx + R[4] * ty + R[5] * tz;
    tout[i * 3 + 2] = trans[i * 3 + 2] + R[6] * tx + R[7] * ty + R[8] * tz;
}

// =====================================================================
// Fused edge MLP: per dst node (32 contiguous edges).
// ein = [z | ne[src] | ne[dst]] (32x384) built in LDS (ne[dst] shared!).
// he = relu(ein@We1+be1) -> LDS ; out = he@We2+be2 -> LN -> z_new.
// z read once, z_new written once: the bandwidth-critical path.
// Static LDS: 14336 floats (56KB) + small arrays. All staging B128.
// =====================================================================
__global__ __launch_bounds__(256) void edge_mlp_kernel(
    const float* __restrict__ z, const float* __restrict__ ne,
    const int* __restrict__ ei, const float* __restrict__ We1,
    const float* __restrict__ be1, const float* __restrict__ We2,
    const float* __restrict__ be2, const float* __restrict__ lng,
    const float* __restrict__ lnb, float* __restrict__ zout)
{
    __shared__ __align__(16) float smem[32 * 384 + 16 * 128]; // ein(48K)+Wslice(8K)
    __shared__ int srcs[32];
    __shared__ float sMu[32], sRs[32];
    float* sEin = smem;
    float* sW   = smem + 32 * 384;
    float* sHe  = smem;             // overlays ein in phase 2
    float* sOut = smem + 32 * 128;  // overlays ein (disjoint from sHe)

    const int n = blockIdx.x, tid = threadIdx.x;
    const int lane = tid & 31, wave = tid >> 5;
    if (tid < 32) srcs[tid] = ei[n * DEG + tid];
    __syncthreads();

    // build ein tile: 32 rows x 96 float4 (all three segments 16B aligned)
    for (int idx = tid; idx < 32 * 96; idx += 256) {
        int rr = idx / 96, cc = (idx - rr * 96) * 4;
        float4 val;
        if (cc < 128)      val = *(const float4*)(z + (size_t)(n * DEG + rr) * CZ + cc);
        else if (cc < 256) val = *(const float4*)(ne + (size_t)srcs[rr] * 128 + (cc - 128));
        else               val = *(const float4*)(ne + (size_t)n * 128 + (cc - 256));
        *(float4*)&sEin[rr * 384 + cc] = val;
    }
    __syncthreads();

    const int col = wave * 16 + (lane & 15); // output col 0..127
    const int kof = (lane >> 4) * 2;
    const int r0 = (lane & 15), r1 = 16 + (lane & 15);

    // GEMM1: 32x384 @ 384x128 (+bias, relu)
    v8f a0 = {}, a1 = {};
    for (int kb = 0; kb < 384; kb += 16) {
#pragma unroll
        for (int i = 0; i < 2; ++i) {          // 512 float4
            int idx = tid + i * 256;
            int r = idx >> 5, c = (idx & 31) * 4;
            *(float4*)&sW[r * 128 + c] =
                *(const float4*)(We1 + (size_t)(kb + r) * 128 + c);
        }
        __syncthreads();
#pragma unroll
        for (int kk = 0; kk < 16; kk += 4) {
            int kc = kk + kof;
            v2f av0, av1, bv;
            bv.x = sW[kc * 128 + col];       bv.y = sW[(kc + 1) * 128 + col];
            av0.x = sEin[r0 * 384 + kb + kc]; av0.y = sEin[r0 * 384 + kb + kc + 1];
            av1.x = sEin[r1 * 384 + kb + kc]; av1.y = sEin[r1 * 384 + kb + kc + 1];
            a0 = __builtin_amdgcn_wmma_f32_16x16x4_f32(false, av0, false, bv, (short)0, a0, false, false);
            a1 = __builtin_amdgcn_wmma_f32_16x16x4_f32(false, av1, false, bv, (short)0, a1, false, false);
        }
        __syncthreads();
    }
    // he -> LDS (ein no longer needed)
    {
        float bv = be1[col];
#pragma unroll
        for (int rr = 0; rr < 8; ++rr) {
            int m = (lane >> 4) * 8 + rr;
            sHe[m * 128 + col]        = fmaxf(a0[rr] + bv, 0.f);
            sHe[(16 + m) * 128 + col] = fmaxf(a1[rr] + bv, 0.f);
        }
    }
    __syncthreads();

    // GEMM2: 32x128 @ 128x128 (+bias)
    v8f c0 = {}, c1 = {};
    for (int kb = 0; kb < 128; kb += 16) {
#pragma unroll
        for (int i = 0; i < 2; ++i) {
            int idx = tid + i * 256;
            int r = idx >> 5, c = (idx & 31) * 4;
            *(float4*)&sW[r * 128 + c] =
                *(const float4*)(We2 + (size_t)(kb + r) * 128 + c);
        }
        __syncthreads();
#pragma unroll
        for (int kk = 0; kk < 16; kk += 4) {
            int kc = kk + kof;
            v2f av0, av1, bv;
            bv.x = sW[kc * 128 + col];       bv.y = sW[(kc + 1) * 128 + col];
            av0.x = sHe[r0 * 128 + kb + kc]; av0.y = sHe[r0 * 128 + kb + kc + 1];
            av1.x = sHe[r1 * 128 + kb + kc]; av1.y = sHe[r1 * 128 + kb + kc + 1];
            c0 = __builtin_amdgcn_wmma_f32_16x16x4_f32(false, av0, false, bv, (short)0, c0, false, false);
            c1 = __builtin_amdgcn_wmma_f32_16x16x4_f32(false, av1, false, bv, (short)0, c1, false, false);
        }
        __syncthreads();
    }
    {
        float bv = be2[col];
#pragma unroll
        for (int rr = 0; rr < 8; ++rr) {
            int m = (lane >> 4) * 8 + rr;
            sOut[m * 128 + col]        = c0[rr] + bv;
            sOut[(16 + m) * 128 + col] = c1[rr] + bv;
        }
    }
    __syncthreads();

    // row LayerNorm (CZ=128)
    if (tid < 32) {
        float s = 0.f;
        for (int cc = 0; cc < 128; ++cc) s += sOut[tid * 128 + cc];
        float mu = s * (1.f / 128.f);
        float vv = 0.f;
        for (int cc = 0; cc < 128; ++cc) { float d = sOut[tid * 128 + cc] - mu; vv += d * d; }
        sMu[tid] = mu;
        sRs[tid] = rsqrtf(vv * (1.f / 128.f) + 1e-5f);
    }
    __syncthreads();
    // B128 store of z_new
    for (int idx = tid; idx < 32 * 32; idx += 256) {
        int rr = idx >> 5, cc = (idx & 31) * 4;
        float mu = sMu[rr], rs = sRs[rr];
        const float* o = &sOut[rr * 128 + cc];
        float4 y;
        y.x = (o[0] - mu) * rs * lng[cc + 0] + lnb[cc + 0];
        y.y = (o[1] - mu) * rs * lng[cc + 1] + lnb[cc + 1];
        y.z = (o[2] - mu) * rs * lng[cc + 2] + lnb[cc + 2];
        y.w = (o[3] - mu) * rs * lng[cc + 3] + lnb[cc + 3];
        *(float4*)(zout + (size_t)(n * DEG + rr) * CZ + cc) = y;
    }
}

// =====================================================================
// host launcher
// =====================================================================
extern "C" void kernel_launch(void* const* d_in, const int* in_sizes, int n_in,
                              void* d_out, int out_size, void* d_ws, size_t ws_size,
                              hipStream_t stream)
{
    (void)in_sizes; (void)n_in; (void)out_size; (void)ws_size;
    const float* s      = (const float*)d_in[0];
    const float* z      = (const float*)d_in[1];
    const int*   ei     = (const int*)d_in[2];   // src rows: ei[0..E)
    const float* quats  = (const float*)d_in[3];
    const float* trans  = (const float*)d_in[4];
    const float* rmask  = (const float*)d_in[5];
    const float* nmask  = (const float*)d_in[6];
    const float* Wq     = (const float*)d_in[7];
    const float* Wk     = (const float*)d_in[8];
    const float* Wv     = (const float*)d_in[9];
    const float* Wqp    = (const float*)d_in[10];
    const float* Wkp    = (const float*)d_in[11];
    const float* Wvp    = (const float*)d_in[12];
    const float* Wb     = (const float*)d_in[13];
    const float* headw  = (const float*)d_in[14];
    const float* Wo     = (const float*)d_in[15];
    const float* bo     = (const float*)d_in[16];
    const float* ln1g   = (const float*)d_in[17];
    const float* ln1b   = (const float*)d_in[18];
    const float* Wt1    = (const float*)d_in[19];
    const float* bt1    = (const float*)d_in[20];
    const float* Wt2    = (const float*)d_in[21];
    const float* bt2    = (const float*)d_in[22];
    const float* Wt3    = (const float*)d_in[23];
    const float* bt3    = (const float*)d_in[24];
    const float* lntg   = (const float*)d_in[25];
    const float* lntb   = (const float*)d_in[26];
    const float* Wbb    = (const float*)d_in[27];
    const float* bbb    = (const float*)d_in[28];
    const float* Wne    = (const float*)d_in[29];
    const float* bne    = (const float*)d_in[30];
    const float* We1    = (const float*)d_in[31];
    const float* be1    = (const float*)d_in[32];
    const float* We2    = (const float*)d_in[33];
    const float* be2    = (const float*)d_in[34];
    const float* lneg   = (const float*)d_in[35];
    const float* lneb   = (const float*)d_in[36];

    float* out = (float*)d_out;
    float* ws  = (float*)d_ws;

    // ---- workspace layout (floats) ----
    float* rot  = ws;                    // N*9
    float* qn   = rot  + NN * 9;         // N*4
    float* qb   = qn   + NN * 4;         // N x 256
    float* kb   = qb   + NN * 256;       // N x 256
    float* vb   = kb   + NN * 256;       // N x 256
    float* qg   = vb   + NN * 256;       // N x 192 (globalized in place)
    float* kg   = qg   + NN * 192;       // N x 192
    float* vg   = kg   + NN * 192;       // N x 288
    float* bws  = vg   + NN * 288;       // E x 8
    float* attw = bws  + (size_t)EE * 8; // E x 8
    float* cat  = attw + (size_t)EE * 8; // N x 1664
    float* s2m  = cat  + (size_t)NN * 1664; // N x 256 (dedicated: no aliasing)
    // regions reused only after their producers/consumers are done:
    float* ipa  = qb;                    // N x 256  (q dead after attn)
    float* s1   = kb;                    // N x 256  (k dead after attn)
    float* h1   = vb;                    // N x 256  (v dead after aggregate)
    float* h2   = qg;                    // N x 256  (qg+kg region, dead after attn)
    float* h3   = qg + NN * 256;         // N x 256  (inside qg..vg = N*672, dead)
    float* updw = bws;                   // N x 6    (bws dead after attn)
    float* ne   = attw;                  // N x 128  (attw dead after aggregate)

    float* s2out = out;                      // N x 256
    float* qout  = s2out + (size_t)NN * 256; // N x 4
    float* tout  = qout + (size_t)NN * 4;    // N x 3
    float* zout  = tout + (size_t)NN * 3;    // E x 128

    dim3 blk(256);
    auto gemm = [&](const float* A, const float* W, const float* bias, float* O,
                    int M, int K, int N, int relu) {
        dim3 grid((N + 63) / 64, (M + 31) / 32);
        wmma_gemm_f32<<<grid, blk, 0, stream>>>(A, W, bias, O, M, K, N, relu);
    };

    node_prep<<<(NN + 255) / 256, blk, 0, stream>>>(quats, qn, rot);

    gemm(s, Wq,  nullptr, qb, NN, CS, H * C, 0);
    gemm(s, Wk,  nullptr, kb, NN, CS, H * C, 0);
    gemm(s, Wv,  nullptr, vb, NN, CS, H * C, 0);
    gemm(s, Wqp, nullptr, qg, NN, CS, H * PQ * 3, 0);
    gemm(s, Wkp, nullptr, kg, NN, CS, H * PQ * 3, 0);
    gemm(s, Wvp, nullptr, vg, NN, CS, H * PV * 3, 0);

    transform_points<<<(NN * H * PQ + 255) / 256, blk, 0, stream>>>(qg, rot, trans, H * PQ);
    transform_points<<<(NN * H * PQ + 255) / 256, blk, 0, stream>>>(kg, rot, trans, H * PQ);
    transform_points<<<(NN * H * PV + 255) / 256, blk, 0, stream>>>(vg, rot, trans, H * PV);

    gemm(z, Wb, nullptr, bws, EE, CZ, H, 0);

    attn_kernel<<<NN, blk, 0, stream>>>(qb, kb, qg, kg, bws, headw, rmask, ei, attw);
    aggregate_kernel<<<NN, blk, 0, stream>>>(vb, vg, z, attw, ei, rot, trans, cat);

    gemm(cat, Wo, bo, ipa, NN, 1664, CS, 0);
    fused_ln<<<NN, blk, 0, stream>>>(s, ipa, rmask, ln1g, ln1b, nullptr, nullptr, s1, nullptr);

    gemm(s1, Wt1, bt1, h1, NN, CS, CS, 1);
    gemm(h1, Wt2, bt2, h2, NN, CS, CS, 1);
    gemm(h2, Wt3, bt3, h3, NN, CS, CS, 0);
    fused_ln<<<NN, blk, 0, stream>>>(s1, h3, nullptr, lntg, lntb, rmask, nmask, s2out, s2m);

    gemm(s2m, Wbb, bbb, updw, NN, CS, 6, 0);
    bb_update<<<(NN + 255) / 256, blk, 0, stream>>>(updw, nmask, qn, rot, trans, qout, tout);

    gemm(s2out, Wne, bne, ne, NN, CS, CS / 2, 0);
    edge_mlp_kernel<<<NN, blk, 0, stream>>>(z, ne, ei, We1, be1, We2, be2, lneg, lneb, zout);
}